// CenterDependentPool2D_61692910239962
// MI455X (gfx1250) — compile-verified
//
#include <hip/hip_runtime.h>
#include <stdint.h>

// CenterDependentPool2D for MI455X (gfx1250).
// Per-pixel radius-dependent max pool, k in {2,8,14,20,26} (k = 2+6L), stride 2,
// reflect padding. Tiled; CDNA5 async global->LDS b128 staging; 3-rung max
// pyramid (rows: P pairs + Q quads; cols of V: HP pairs + HQ quads);
// per-tile ring-level restriction; NT output stores.

#define IN_SZ   448
#define OUT_SZ  224
#define CENTER  112
#define TOX     32                  // output tile width
#define TOY     16                  // output tile height
#define IW      (2*TOX + 24)        // 88 input cols incl. halo
#define IH      (2*TOY + 24)        // 56 input rows incl. halo
#define IW4     (IW/4)              // 22 float4 per input row
#define PROWS   (IH/2)              // 28 pair rows
#define QROWS   (IH/4)              // 14 quad rows
#define HPW     (IW/2)              // 44 horizontal pairs per row
#define HQW     (IW/4)              // 22 horizontal quads per row
#define NLEV    5
#define NTHREADS 256

#define NEG_INF (-__builtin_inff())

typedef float v2f __attribute__((ext_vector_type(2)));   // native vector for NT store

__device__ __forceinline__ int reflect448(int i) {
    // jnp.pad "reflect": -p -> p ; (N-1)+p -> (N-1)-p  (pads <= 13, single reflection)
    i = (i < 0) ? -i : i;
    return (i >= IN_SZ) ? (2*IN_SZ - 2 - i) : i;
}
__device__ __forceinline__ float4 max4(float4 a, float4 b) {
    return make_float4(fmaxf(a.x,b.x), fmaxf(a.y,b.y), fmaxf(a.z,b.z), fmaxf(a.w,b.w));
}
__device__ __forceinline__ int level_of(int d2) {
    // strict '<' disk boundaries at radii 60,75,90,105
    return (d2 >= 3600) + (d2 >= 5625) + (d2 >= 8100) + (d2 >= 11025);
}
__device__ __forceinline__ int iabs_(int a)        { return a < 0 ? -a : a; }
__device__ __forceinline__ int imin_(int a, int b) { return a < b ? a : b; }
__device__ __forceinline__ int imax_(int a, int b) { return a > b ? a : b; }

__global__ __launch_bounds__(NTHREADS)
void cdp2d_kernel(const float* __restrict__ x, float* __restrict__ out) {
    __shared__ float4 tile4[IH * IW4];                 // 19.7 KB staged input
    __shared__ float4 P4[PROWS * IW4];                 //  9.9 KB row-pair maxes
    __shared__ float4 Q4[QROWS * IW4];                 //  4.9 KB row-quad maxes
    __shared__ float4 V4[NLEV * TOY * IW4];            // 28.2 KB per-level vertical maxes
    __shared__ float2 HP2[NLEV * TOY * IW4];           // 14.1 KB col pairs of V
    __shared__ float  HQ[NLEV * TOY * HQW];            //  7.0 KB col quads of V
    float* V  = (float*)V4;
    float* HP = (float*)HP2;

    const int tid = threadIdx.x;
    const int ox0 = blockIdx.x * TOX;                  // 7 tiles in x
    const int oy0 = blockIdx.y * TOY;                  // 14 tiles in y
    const int bc  = blockIdx.z;                        // B*C images

    const int rowbase = 2*oy0 - 12;
    const int colbase = 2*ox0 - 12;
    const float* src = x + (size_t)bc * (IN_SZ * IN_SZ);

    // ---- uniform per-tile ring-level range [Lmin, Lmax] ----
    const int ax0 = ox0 - CENTER, ax1 = ox0 + TOX - 1 - CENTER;
    const int ay0 = oy0 - CENTER, ay1 = oy0 + TOY - 1 - CENTER;
    const int dxmin = (ax0 <= 0 && ax1 >= 0) ? 0 : imin_(iabs_(ax0), iabs_(ax1));
    const int dymin = (ay0 <= 0 && ay1 >= 0) ? 0 : imin_(iabs_(ay0), iabs_(ay1));
    const int dxmax = imax_(iabs_(ax0), iabs_(ax1));
    const int dymax = imax_(iabs_(ay0), iabs_(ay1));
    const int Lmin  = level_of(dxmin*dxmin + dymin*dymin);
    const int Lmax  = level_of(dxmax*dxmax + dymax*dymax);

    // ---- Stage 0: async copy of (reflect-padded) input tile into LDS ----
    const bool interior = (rowbase >= 0) && (rowbase + IH <= IN_SZ) &&
                          (colbase >= 0) && (colbase + IW <= IN_SZ);
    const uint32_t lds0 = (uint32_t)(uintptr_t)&tile4[0];   // low 32 bits = LDS byte addr
    if (interior) {
        // fast path: strided pointer march, pure b128 DMA, no reflect math
        int c4 = tid % IW4;                            // computed once
        int r0 = tid / IW4;                            // 0..11 ; tid >= 242 idle
        if (tid < 11 * IW4) {                          // 242 active: 11 rows x 22 vec4 per pass
            const float* gp = src + (size_t)(rowbase + r0) * IN_SZ + colbase + 4 * c4;
            uint32_t lds = lds0 + (uint32_t)(r0 * IW4 + c4) * 16u;
            for (int r = r0; r < IH; r += 11) {
                uint64_t ga = (uint64_t)(uintptr_t)gp;
                asm volatile("global_load_async_to_lds_b128 %0, %1, off"
                             :: "v"(lds), "v"(ga) : "memory");
                gp  += 11 * IN_SZ;
                lds += 11 * IW4 * 16;
            }
        }
    } else {
        // border tiles: generic path with reflect index mapping
        for (int p = tid; p < IH * IW4; p += NTHREADS) {
            int r  = p / IW4;
            int c4 = p - r * IW4;
            int gr = reflect448(rowbase + r);          // row reflect keeps rows contiguous
            int gc0 = colbase + 4 * c4;
            const float* gp = src + (size_t)gr * IN_SZ;
            uint32_t lds = lds0 + (uint32_t)p * 16u;
            if (gc0 >= 0 && gc0 + 3 < IN_SZ) {         // contiguous 16B
                uint64_t ga = (uint64_t)(uintptr_t)(gp + gc0);
                asm volatile("global_load_async_to_lds_b128 %0, %1, off"
                             :: "v"(lds), "v"(ga) : "memory");
            } else {                                   // column-reflect edge: 4x b32
                #pragma unroll
                for (int e = 0; e < 4; ++e) {
                    int gc = reflect448(gc0 + e);
                    uint64_t ga = (uint64_t)(uintptr_t)(gp + gc);
                    uint32_t l2 = lds + 4u * e;
                    asm volatile("global_load_async_to_lds_b32 %0, %1, off"
                                 :: "v"(l2), "v"(ga) : "memory");
                }
            }
        }
    }
    asm volatile("s_wait_asynccnt 0" ::: "memory");    // this wave's async LDS writes landed
    __syncthreads();

    // ---- Stage PQ (fused): row-pair maxes P and row-quad maxes Q ----
    for (int p = tid; p < QROWS * IW4; p += NTHREADS) { // 308 positions
        int j  = p / IW4;
        int c4 = p - j * IW4;
        float4 r0 = tile4[(4*j    ) * IW4 + c4];
        float4 r1 = tile4[(4*j + 1) * IW4 + c4];
        float4 r2 = tile4[(4*j + 2) * IW4 + c4];
        float4 r3 = tile4[(4*j + 3) * IW4 + c4];
        float4 p0 = max4(r0, r1);
        float4 p1 = max4(r2, r3);
        P4[(2*j    ) * IW4 + c4] = p0;
        P4[(2*j + 1) * IW4 + c4] = p1;
        Q4[j * IW4 + c4] = max4(p0, p1);
    }
    __syncthreads();

    // ---- Stage V: per-level vertical max via singles/pairs/quads (+ fused HP, HQ) ----
    // Level-L window rows: [2*oy+12-3L, 2*oy+14+3L)
    for (int p = tid; p < TOY * IW4; p += NTHREADS) {   // 352 positions
        int oy = p / IW4;
        int c4 = p - oy * IW4;
        for (int L = Lmin; L <= Lmax; ++L) {
            int a = 2*oy + 12 - 3*L;
            int b = a + 2 + 6*L;
            float4 v = make_float4(NEG_INF, NEG_INF, NEG_INF, NEG_INF);
            if (a & 1) { v = max4(v, tile4[a * IW4 + c4]); a++; }
            if (b & 1) { b--; v = max4(v, tile4[b * IW4 + c4]); }
            int pa = a >> 1, pb = b >> 1;               // pair range [pa, pb)
            if (pa & 1) { v = max4(v, P4[pa * IW4 + c4]); pa++; }
            if (pb & 1) { pb--; v = max4(v, P4[pb * IW4 + c4]); }
            for (int q = (pa >> 1); q < (pb >> 1); ++q) // quad range
                v = max4(v, Q4[q * IW4 + c4]);
            V4[(L*TOY + oy) * IW4 + c4] = v;
            float2 hp = make_float2(fmaxf(v.x, v.y), fmaxf(v.z, v.w));
            HP2[(L*TOY + oy) * IW4 + c4] = hp;
            HQ [(L*TOY + oy) * HQW + c4] = fmaxf(hp.x, hp.y);
        }
    }
    __syncthreads();

    // ---- Stage 2: per-pixel horizontal max via singles/pairs/quads; 2 outputs/thread ----
    {
        int oy  = tid >> 4;                            // 0..15
        int oxp = (tid & 15) << 1;                     // 0,2,...,30
        int oyg = oy0 + oy;
        int dy  = oyg - CENTER;
        v2f res;
        #pragma unroll
        for (int e = 0; e < 2; ++e) {
            int ox  = oxp + e;
            int oxg = ox0 + ox;
            int dx  = oxg - CENTER;
            int L   = level_of(dx*dx + dy*dy);
            int a   = 2*ox + 12 - 3*L;                 // window cols [a, b)
            int b   = a + 2 + 6*L;
            const float* Vrow  = V  + (L*TOY + oy) * IW;
            const float* HProw = HP + (L*TOY + oy) * HPW;
            const float* HQrow = HQ + (L*TOY + oy) * HQW;
            float m = NEG_INF;
            if (a & 1) { m = fmaxf(m, Vrow[a]); a++; }
            if (b & 1) { b--; m = fmaxf(m, Vrow[b]); }
            int pa = a >> 1, pb = b >> 1;
            if (pa & 1) { m = fmaxf(m, HProw[pa]); pa++; }
            if (pb & 1) { pb--; m = fmaxf(m, HProw[pb]); }
            for (int j = (pa >> 1); j < (pb >> 1); ++j)
                m = fmaxf(m, HQrow[j]);
            res[e] = m;
        }
        v2f* orow = (v2f*)(out + ((size_t)bc * OUT_SZ + oyg) * OUT_SZ + ox0 + oxp);
        __builtin_nontemporal_store(res, orow);        // coalesced 8B NT store
    }
}

extern "C" void kernel_launch(void* const* d_in, const int* in_sizes, int n_in,
                              void* d_out, int out_size, void* d_ws, size_t ws_size,
                              hipStream_t stream) {
    const float* x = (const float*)d_in[0];
    float* out = (float*)d_out;
    int bc_total = in_sizes[0] / (IN_SZ * IN_SZ);      // B*C = 512
    dim3 grid(OUT_SZ / TOX, OUT_SZ / TOY, bc_total);
    dim3 block(NTHREADS);
    hipLaunchKernelGGL(cdp2d_kernel, grid, block, 0, stream, x, out);
}